// GATNet_39711267619290
// MI455X (gfx1250) — compile-verified
//
#include <hip/hip_runtime.h>
#include <math.h>

// ---------------------------------------------------------------- constants
constexpr int Nn    = 50000;    // nodes
constexpr int Ee    = 800000;   // edges (before self loops)
constexpr int F_IN  = 256;
constexpr int HID   = 64;
constexpr int NCLS  = 40;
constexpr int HEADS = 8;
constexpr int ETOT  = Ee + Nn;  // edges + self loops
constexpr float NEG_SLOPE = 0.2f;

typedef float v2f __attribute__((ext_vector_type(2)));
typedef float v8f __attribute__((ext_vector_type(8)));

// ---------------------------------------------------------------- utilities
__global__ void zero_f32(float* __restrict__ p, size_t n) {
  size_t i = (size_t)blockIdx.x * blockDim.x + threadIdx.x;
  if (i < n) p[i] = 0.0f;
}

// Monotone float<->uint mapping so atomicMax(uint) == float max.
__device__ __forceinline__ unsigned flipf(float f) {
  unsigned u = __float_as_uint(f);
  return (u & 0x80000000u) ? ~u : (u | 0x80000000u);
}
__device__ __forceinline__ float unflipf(unsigned u) {
  return __uint_as_float((u & 0x80000000u) ? (u ^ 0x80000000u) : ~u);
}

// ---------------------------------------------------------------- WMMA GEMM
// Out[M, NOUT] = A[M, K] @ W[NOUT, K]^T   (row-major A and W, PyG linear).
// Block = 4 waves; each wave owns a 16-row M tile, all waves share one
// 64-column W strip (WGP$-resident). Per wave: 4 accumulators (16x64 strip),
// K-loop software-pipelined (next fragments loaded before current WMMAs so
// the 4 v_wmma_f32_16x16x4_f32 per step overlap the 5 in-flight loads).
// f32 16x16x4 layout: lanes 0-15 hold K=kb+{0,1}, lanes 16-31 K=kb+{2,3};
// A rows / B cols indexed by (lane & 15). C/D: n = lane&15, m = v+8*(lane>>4).
__global__ __launch_bounds__(128) void gemm16x64_wmma_f32(
    const float* __restrict__ A, const float* __restrict__ W,
    float* __restrict__ Out, int K, int NOUT) {
  const int lane = threadIdx.x & 31;
  const int wave = threadIdx.x >> 5;
  const int mb = (blockIdx.x * 4 + wave) * 16;
  if (mb >= Nn) return;                      // whole-tile guard (full 16-row tiles)
  const int nb = blockIdx.y * 64;
  const int r  = lane & 15;
  const int kh = (lane >> 4) << 1;           // 0 or 2

  const float* ap = A + (size_t)(mb + r) * K + kh;
  const float* wp = W + (size_t)(nb + r) * K + kh;
  const size_t ws16 = (size_t)16 * K;        // 16 output columns ahead in W

  v8f acc0 = v8f{0}, acc1 = v8f{0}, acc2 = v8f{0}, acc3 = v8f{0};

  // prologue: fragments for k = 0
  v2f a  = *(const v2f*)(ap);
  v2f b0 = *(const v2f*)(wp);
  v2f b1 = *(const v2f*)(wp + ws16);
  v2f b2 = *(const v2f*)(wp + 2 * ws16);
  v2f b3 = *(const v2f*)(wp + 3 * ws16);

#pragma unroll 4
  for (int k = 4; k < K; k += 4) {
    // issue next step's loads before consuming current fragments
    v2f an  = *(const v2f*)(ap + k);
    v2f bn0 = *(const v2f*)(wp + k);
    v2f bn1 = *(const v2f*)(wp + ws16 + k);
    v2f bn2 = *(const v2f*)(wp + 2 * ws16 + k);
    v2f bn3 = *(const v2f*)(wp + 3 * ws16 + k);

    acc0 = __builtin_amdgcn_wmma_f32_16x16x4_f32(false, a, false, b0, (short)0, acc0, false, false);
    acc1 = __builtin_amdgcn_wmma_f32_16x16x4_f32(false, a, false, b1, (short)0, acc1, false, false);
    acc2 = __builtin_amdgcn_wmma_f32_16x16x4_f32(false, a, false, b2, (short)0, acc2, false, false);
    acc3 = __builtin_amdgcn_wmma_f32_16x16x4_f32(false, a, false, b3, (short)0, acc3, false, false);

    a = an; b0 = bn0; b1 = bn1; b2 = bn2; b3 = bn3;
  }
  // epilogue: last k step
  acc0 = __builtin_amdgcn_wmma_f32_16x16x4_f32(false, a, false, b0, (short)0, acc0, false, false);
  acc1 = __builtin_amdgcn_wmma_f32_16x16x4_f32(false, a, false, b1, (short)0, acc1, false, false);
  acc2 = __builtin_amdgcn_wmma_f32_16x16x4_f32(false, a, false, b2, (short)0, acc2, false, false);
  acc3 = __builtin_amdgcn_wmma_f32_16x16x4_f32(false, a, false, b3, (short)0, acc3, false, false);

  const int mrow = mb + ((lane >> 4) << 3);
  const int ncol = nb + (lane & 15);
  float* o0 = Out + (size_t)mrow * NOUT + ncol;
#pragma unroll
  for (int v = 0; v < 8; ++v) {
    float* orow = o0 + (size_t)v * NOUT;
    orow[0]  = acc0[v];
    orow[16] = acc1[v];
    orow[32] = acc2[v];
    orow[48] = acc3[v];
  }
}

// ---------------------------------------------------------------- attention logits
// a_s[n,h] = dot(hfeat[n,h,:], att_src[h,:]); same for a_d. One thread per (n,h).
template <int C>
__global__ __launch_bounds__(256) void attn_logits(
    const float* __restrict__ hfeat, const float* __restrict__ att_s,
    const float* __restrict__ att_d, float* __restrict__ a_s,
    float* __restrict__ a_d) {
  int t = blockIdx.x * blockDim.x + threadIdx.x;    // t = n*HEADS + h
  if (t >= Nn * HEADS) return;
  int h = t & (HEADS - 1);
  const float* hp = hfeat + (size_t)t * C;
  const float* wsp = att_s + h * C;
  const float* wdp = att_d + h * C;
  float ss = 0.f, sd = 0.f;
#pragma unroll 8
  for (int c = 0; c < C; ++c) {
    float v = hp[c];
    ss += v * wsp[c];
    sd += v * wdp[c];
  }
  a_s[t] = ss;
  a_d[t] = sd;
}

// ---------------------------------------------------------------- edge passes
__device__ __forceinline__ void edge_nodes(const int* __restrict__ ei, int e,
                                           int& src, int& dst) {
  if (e < Ee) { src = ei[e]; dst = ei[Ee + e]; }
  else        { src = e - Ee; dst = e - Ee; }       // appended self loops
}
__device__ __forceinline__ float leaky(float v) {
  return v > 0.f ? v : NEG_SLOPE * v;
}

// Pass 1: segment max (in flipped-uint space; init 0 is below any real value,
// and every dst has at least its self loop).
__global__ __launch_bounds__(256) void edge_max(
    const int* __restrict__ ei, const float* __restrict__ a_s,
    const float* __restrict__ a_d, unsigned* __restrict__ m) {
  int e = blockIdx.x * blockDim.x + threadIdx.x;
  if (e >= ETOT) return;
  int src, dst; edge_nodes(ei, e, src, dst);
#pragma unroll
  for (int h = 0; h < HEADS; ++h) {
    float v = leaky(a_s[src * HEADS + h] + a_d[dst * HEADS + h]);
    atomicMax(&m[dst * HEADS + h], flipf(v));
  }
}

// Pass 2: denom[dst,h] += exp(e - m[dst,h])
__global__ __launch_bounds__(256) void edge_denom(
    const int* __restrict__ ei, const float* __restrict__ a_s,
    const float* __restrict__ a_d, const unsigned* __restrict__ m,
    float* __restrict__ denom) {
  int e = blockIdx.x * blockDim.x + threadIdx.x;
  if (e >= ETOT) return;
  int src, dst; edge_nodes(ei, e, src, dst);
#pragma unroll
  for (int h = 0; h < HEADS; ++h) {
    float v = leaky(a_s[src * HEADS + h] + a_d[dst * HEADS + h]);
    float p = __expf(v - unflipf(m[dst * HEADS + h]));
    atomicAdd(&denom[dst * HEADS + h], p);
  }
}

// Pass 3: out[dst,c] += sum_h (alpha[e,h]/HEADS) * hfeat[src,h,c]
// (head-mean folded in: mean over heads is linear). One wave per edge.
template <int C>
__global__ __launch_bounds__(256) void edge_aggregate(
    const int* __restrict__ ei, const float* __restrict__ a_s,
    const float* __restrict__ a_d, const unsigned* __restrict__ m,
    const float* __restrict__ denom, const float* __restrict__ hfeat,
    float* __restrict__ out) {
  int gt = blockIdx.x * blockDim.x + threadIdx.x;
  int e = gt >> 5;
  int lane = threadIdx.x & 31;
  if (e >= ETOT) return;
  int src, dst; edge_nodes(ei, e, src, dst);

  float coef = 0.f;
  if (lane < HEADS) {
    float v = leaky(a_s[src * HEADS + lane] + a_d[dst * HEADS + lane]);
    float p = __expf(v - unflipf(m[dst * HEADS + lane]));
    coef = p / (denom[dst * HEADS + lane] + 1e-16f) * (1.0f / HEADS);
  }
  float c0 = __shfl(coef, 0), c1 = __shfl(coef, 1), c2 = __shfl(coef, 2),
        c3 = __shfl(coef, 3), c4 = __shfl(coef, 4), c5 = __shfl(coef, 5),
        c6 = __shfl(coef, 6), c7 = __shfl(coef, 7);

  const float* hp = hfeat + (size_t)src * HEADS * C;
  float* op = out + (size_t)dst * C;
#pragma unroll
  for (int c = lane; c < C; c += 32) {
    float s = c0 * hp[c]         + c1 * hp[C + c]     + c2 * hp[2 * C + c] +
              c3 * hp[3 * C + c] + c4 * hp[4 * C + c] + c5 * hp[5 * C + c] +
              c6 * hp[6 * C + c] + c7 * hp[7 * C + c];
    atomicAdd(&op[c], s);
  }
}

// ---------------------------------------------------------------- epilogues
__global__ __launch_bounds__(256) void bias_relu(
    float* __restrict__ h1, const float* __restrict__ b0) {
  int t = blockIdx.x * blockDim.x + threadIdx.x;
  if (t >= Nn * HID) return;
  float v = h1[t] + b0[t & (HID - 1)];
  h1[t] = v > 0.f ? v : 0.f;
}

// One wave per node: z = z2 + b2; out = z - logsumexp(z). 40 classes:
// lane holds z[lane], lanes 0..7 also hold z[32+lane].
__global__ __launch_bounds__(256) void bias_logsoftmax(
    const float* __restrict__ z2, const float* __restrict__ b2,
    float* __restrict__ out) {
  int gt = blockIdx.x * blockDim.x + threadIdx.x;
  int n = gt >> 5;
  int lane = threadIdx.x & 31;
  if (n >= Nn) return;
  const float* zp = z2 + (size_t)n * NCLS;
  float z0 = zp[lane] + b2[lane];
  float z1 = (lane < NCLS - 32) ? (zp[32 + lane] + b2[32 + lane]) : -1e30f;
  float mx = fmaxf(z0, z1);
#pragma unroll
  for (int o = 16; o; o >>= 1) mx = fmaxf(mx, __shfl_xor(mx, o));
  float s = __expf(z0 - mx) + ((lane < NCLS - 32) ? __expf(z1 - mx) : 0.f);
#pragma unroll
  for (int o = 16; o; o >>= 1) s += __shfl_xor(s, o);
  float lse = __logf(s) + mx;
  float* op = out + (size_t)n * NCLS;
  op[lane] = z0 - lse;
  if (lane < NCLS - 32) op[32 + lane] = z1 - lse;
}

// ---------------------------------------------------------------- launch
static inline int cdiv(long long a, long long b) { return (int)((a + b - 1) / b); }

extern "C" void kernel_launch(void* const* d_in, const int* in_sizes, int n_in,
                              void* d_out, int out_size, void* d_ws, size_t ws_size,
                              hipStream_t stream) {
  const float* x   = (const float*)d_in[0];
  const int*   ei  = (const int*)d_in[1];
  // d_in[2] = edge_attr: unused by GATConv (matches reference)
  const float* W0  = (const float*)d_in[3];
  const float* a0s = (const float*)d_in[4];
  const float* a0d = (const float*)d_in[5];
  const float* b0  = (const float*)d_in[6];
  const float* W2  = (const float*)d_in[7];
  const float* a2s = (const float*)d_in[8];
  const float* a2d = (const float*)d_in[9];
  const float* b2  = (const float*)d_in[10];
  float* out = (float*)d_out;

  // ---- workspace carve-up (floats). Total ~32.4M floats (~130 MB).
  float* ws = (float*)d_ws;
  float* hfeat = ws;                                   // [N,8,64] (reused [N,8,40])
  float* h1    = hfeat + (size_t)Nn * HEADS * HID;     // [N,64]  layer-1 output
  float* z2    = h1 + (size_t)Nn * HID;                // [N,40]  layer-2 pre-softmax
  float* a_s   = z2 + (size_t)Nn * NCLS;               // [N,8]
  float* a_d   = a_s + (size_t)Nn * HEADS;             // [N,8]
  unsigned* m  = (unsigned*)(a_d + (size_t)Nn * HEADS);// [N,8] flipped-uint max
  float* denom = (float*)(m + (size_t)Nn * HEADS);     // [N,8]

  const int B = 256;
  const int mblocks = cdiv(Nn, 64);                    // 4 M-tiles (waves) per block

  // ======================= Layer 1 (heads=8, C=HID=64) =======================
  zero_f32<<<cdiv((long long)Nn * HEADS, B), B, 0, stream>>>((float*)m, (size_t)Nn * HEADS);
  zero_f32<<<cdiv((long long)Nn * HEADS, B), B, 0, stream>>>(denom, (size_t)Nn * HEADS);
  zero_f32<<<cdiv((long long)Nn * HID, B), B, 0, stream>>>(h1, (size_t)Nn * HID);

  // hfeat = x @ W0^T : [50000,256] x [512,256]^T, 16x64 strip per wave
  gemm16x64_wmma_f32<<<dim3(mblocks, (HEADS * HID) / 64), 128, 0, stream>>>(
      x, W0, hfeat, F_IN, HEADS * HID);

  attn_logits<HID><<<cdiv((long long)Nn * HEADS, B), B, 0, stream>>>(hfeat, a0s, a0d, a_s, a_d);

  edge_max<<<cdiv(ETOT, B), B, 0, stream>>>(ei, a_s, a_d, m);
  edge_denom<<<cdiv(ETOT, B), B, 0, stream>>>(ei, a_s, a_d, m, denom);
  edge_aggregate<HID><<<cdiv((long long)ETOT * 32, B), B, 0, stream>>>(
      ei, a_s, a_d, m, denom, hfeat, h1);

  bias_relu<<<cdiv((long long)Nn * HID, B), B, 0, stream>>>(h1, b0);

  // ======================= Layer 2 (heads=8, C=NCLS=40) ======================
  zero_f32<<<cdiv((long long)Nn * HEADS, B), B, 0, stream>>>((float*)m, (size_t)Nn * HEADS);
  zero_f32<<<cdiv((long long)Nn * HEADS, B), B, 0, stream>>>(denom, (size_t)Nn * HEADS);
  zero_f32<<<cdiv((long long)Nn * NCLS, B), B, 0, stream>>>(z2, (size_t)Nn * NCLS);

  // hfeat = h1 @ W2^T : [50000,64] x [320,64]^T
  gemm16x64_wmma_f32<<<dim3(mblocks, (HEADS * NCLS) / 64), 128, 0, stream>>>(
      h1, W2, hfeat, HID, HEADS * NCLS);

  attn_logits<NCLS><<<cdiv((long long)Nn * HEADS, B), B, 0, stream>>>(hfeat, a2s, a2d, a_s, a_d);

  edge_max<<<cdiv(ETOT, B), B, 0, stream>>>(ei, a_s, a_d, m);
  edge_denom<<<cdiv(ETOT, B), B, 0, stream>>>(ei, a_s, a_d, m, denom);
  edge_aggregate<NCLS><<<cdiv((long long)ETOT * 32, B), B, 0, stream>>>(
      ei, a_s, a_d, m, denom, hfeat, z2);

  bias_logsoftmax<<<cdiv((long long)Nn * 32, B), B, 0, stream>>>(z2, b2, out);
}